// Renderer_80444737454430
// MI455X (gfx1250) — compile-verified
//
#include <hip/hip_runtime.h>
#include <hip/hip_bf16.h>
#include <math.h>

// Problem constants (fixed by the reference).
#define NUM_RAYS  8192
#define P         1024
#define TILE      128                  // points per wave-tile (4 per lane * 32 lanes)
#define NTILES    (P / TILE)           // 8
#define WPB       8                    // waves (rays) per block
#define THREADS   (WPB * 32)

// ---- gfx1250 async global->LDS path (guarded so we always compile) --------
#if defined(__gfx1250__) && __has_builtin(__builtin_amdgcn_global_load_async_to_lds_b128)
#define USE_ASYNC_LDS 1
#else
#define USE_ASYNC_LDS 0
#endif

#if USE_ASYNC_LDS
#if __has_builtin(__builtin_amdgcn_s_wait_asynccnt)
#define ASYNC_WAIT(n) __builtin_amdgcn_s_wait_asynccnt(n)
#else
#define ASYNC_WAIT(n) asm volatile("s_wait_asynccnt " #n ::: "memory")
#endif
// Builtin signature (from hipcc diagnostic): params are pointers to
// int __attribute__((vector_size(16))) in global (AS1) / LDS (AS3) space.
typedef int v4i __attribute__((vector_size(16)));
typedef __attribute__((address_space(1))) v4i Gv4;
typedef __attribute__((address_space(3))) v4i Lv4;

// Copy 48 contiguous bytes (12 floats = 4 rgb points) global -> LDS, async.
// INST_OFFSET applies to both global and LDS addresses per the ISA.
__device__ __forceinline__ void stage_rgb48(const float* g, float* l) {
  Gv4* gp = (Gv4*)g;
  Lv4* lp = (Lv4*)l;
  __builtin_amdgcn_global_load_async_to_lds_b128(gp, lp, 0, 0);
  __builtin_amdgcn_global_load_async_to_lds_b128(gp, lp, 16, 0);
  __builtin_amdgcn_global_load_async_to_lds_b128(gp, lp, 32, 0);
}
#endif

__global__ __launch_bounds__(THREADS) void compose_rays_kernel(
    const float* __restrict__ rgb,    // [R, P, 3]
    const float* __restrict__ alpha,  // [R, P]
    const float* __restrict__ zv,     // [R, P]
    const float* __restrict__ dirs,   // [R, 3]
    const int* __restrict__ ppr,      // scalar points_per_ray
    float* __restrict__ out)          // [R*3 | R | R | R]
{
#if USE_ASYNC_LDS
  __shared__ __align__(16) float srgb[2][WPB][TILE * 3];  // 24 KB double buffer
#endif
  const int lane = threadIdx.x & 31;
  const int wv   = threadIdx.x >> 5;
  const int ray  = blockIdx.x * WPB + wv;

  const float* A = alpha + (size_t)ray * P;
  const float* Z = zv    + (size_t)ray * P;
  const float* C = rgb   + (size_t)ray * P * 3;

  const float dx = dirs[ray * 3 + 0];
  const float dy = dirs[ray * 3 + 1];
  const float dz = dirs[ray * 3 + 2];
  const float nrm = sqrtf(dx * dx + dy * dy + dz * dz);
  const float cf  = (float)ppr[0] * (1.0f / 1024.0f);   // corrective factor

  float carry = 1.0f;                    // transmittance entering current tile
  float accw = 0.f, r0 = 0.f, r1 = 0.f, r2 = 0.f, dpart = 0.f;
  float wlast = 0.f, zlast = 0.f;        // weight/z of point P-1 (lane 31 only)

#if USE_ASYNC_LDS
  stage_rgb48(C + lane * 12, &srgb[0][wv][lane * 12]);   // prefetch tile 0
#endif

#pragma unroll
  for (int t = 0; t < NTILES; ++t) {
    const int p0 = t * TILE + lane * 4;
    const float4 a4 = *(const float4*)(A + p0);
    const float4 z4 = *(const float4*)(Z + p0);

    // z of point p0+4 (next lane's first z; lane 31 fetches next tile's first)
    float znext = __shfl_down(z4.x, 1);
    const bool is_last = (t == NTILES - 1) && (lane == 31);  // owns point 1023
    if (lane == 31 && t < NTILES - 1) znext = Z[(t + 1) * TILE];

    const float d0 = (z4.y - z4.x) * nrm;
    const float d1 = (z4.z - z4.y) * nrm;
    const float d2 = (z4.w - z4.z) * nrm;
    const float d3 = is_last ? (1e10f * nrm) : ((znext - z4.w) * nrm);

    // alpha_i = 1 - exp(-relu(raw)*dist*cf)
    const float al0 = 1.f - __expf(-fmaxf(a4.x, 0.f) * d0 * cf);
    const float al1 = 1.f - __expf(-fmaxf(a4.y, 0.f) * d1 * cf);
    const float al2 = 1.f - __expf(-fmaxf(a4.z, 0.f) * d2 * cf);
    const float al3 = 1.f - __expf(-fmaxf(a4.w, 0.f) * d3 * cf);

    const float t0 = 1.f - al0 + 1e-10f;
    const float t1 = 1.f - al1 + 1e-10f;
    const float t2 = 1.f - al2 + 1e-10f;
    const float t3 = 1.f - al3 + 1e-10f;

    // wave-level inclusive multiplicative scan of per-lane products
    float s = t0 * t1 * t2 * t3;
#pragma unroll
    for (int off = 1; off < 32; off <<= 1) {
      const float o = __shfl_up(s, off);
      if (lane >= off) s *= o;
    }
    float excl = __shfl_up(s, 1);
    if (lane == 0) excl = 1.f;

    // exclusive transmittance at this lane's first point, then chain locally
    float tr = carry * excl;
    const float w0 = al0 * tr; tr *= t0;
    const float w1 = al1 * tr; tr *= t1;
    const float w2 = al2 * tr; tr *= t2;
    const float w3 = al3 * tr;
    carry *= __shfl(s, 31);              // whole-tile product -> next tile

#if USE_ASYNC_LDS
    if (t + 1 < NTILES) {                // stage next rgb tile, wait for current
      stage_rgb48(C + (size_t)(t + 1) * TILE * 3 + lane * 12,
                  &srgb[(t + 1) & 1][wv][lane * 12]);
      ASYNC_WAIT(3);                     // <=3 outstanding => tile t complete
    } else {
      ASYNC_WAIT(0);
    }
    const float4* cp = (const float4*)(&srgb[t & 1][wv][lane * 12]);
#else
    const float4* cp = (const float4*)(C + (size_t)p0 * 3);
#endif
    const float4 c0 = cp[0];   // p0:{r,g,b} p1:{r}
    const float4 c1 = cp[1];   // p1:{g,b} p2:{r,g}
    const float4 c2 = cp[2];   // p2:{b} p3:{r,g,b}

    accw += (w0 + w1) + (w2 + w3);
    r0 += w0 * c0.x + w1 * c0.w + w2 * c1.z + w3 * c2.y;
    r1 += w0 * c0.y + w1 * c1.x + w2 * c1.w + w3 * c2.z;
    r2 += w0 * c0.z + w1 * c1.y + w2 * c2.x + w3 * c2.w;

    dpart += w0 * z4.x + w1 * z4.y + w2 * z4.z;          // depth excludes p==P-1
    if (is_last) { wlast = w3; zlast = z4.w; }
    else         { dpart += w3 * z4.w; }
  }

  // wave reduction (all lanes end with the totals)
#pragma unroll
  for (int off = 16; off > 0; off >>= 1) {
    accw  += __shfl_xor(accw,  off);
    r0    += __shfl_xor(r0,    off);
    r1    += __shfl_xor(r1,    off);
    r2    += __shfl_xor(r2,    off);
    dpart += __shfl_xor(dpart, off);
  }
  const float wl = __shfl(wlast, 31);
  const float zl = __shfl(zlast, 31);

  if (lane == 0) {
    out[ray * 3 + 0] = r0;
    out[ray * 3 + 1] = r1;
    out[ray * 3 + 2] = r2;
    out[NUM_RAYS * 3 + ray] = accw;
    const float depth = dpart + (1.f - accw + wl) * zl;
    out[NUM_RAYS * 4 + ray] = depth;
    out[NUM_RAYS * 5 + ray] = 1.f / fmaxf(1e-4f, depth / accw);
  }
}

extern "C" void kernel_launch(void* const* d_in, const int* in_sizes, int n_in,
                              void* d_out, int out_size, void* d_ws, size_t ws_size,
                              hipStream_t stream) {
  // setup_inputs order: rgb, raw_alpha, z_vals, rays_dir, pruning_mask(unused),
  //                     points_per_ray (scalar int)
  const float* rgb   = (const float*)d_in[0];
  const float* alpha = (const float*)d_in[1];
  const float* z     = (const float*)d_in[2];
  const float* dirs  = (const float*)d_in[3];
  const int*   ppr   = (const int*)d_in[5];
  float* out = (float*)d_out;
  (void)in_sizes; (void)n_in; (void)d_ws; (void)ws_size; (void)out_size;

  compose_rays_kernel<<<dim3(NUM_RAYS / WPB), dim3(THREADS), 0, stream>>>(
      rgb, alpha, z, dirs, ppr, out);
}